// Quantize_55327768707293
// MI455X (gfx1250) — compile-verified
//
#include <hip/hip_runtime.h>
#include <stdint.h>

constexpr int GN      = 8192;       // matrix dimension
constexpr int TILE    = 128;        // quantization tile (square)
constexpr int NTILES  = GN / TILE;  // 64 tiles per dimension
constexpr int MAXEXP  = 1;

typedef __attribute__((ext_vector_type(2))) float    v2f;
typedef __attribute__((ext_vector_type(4))) float    v4f;
typedef __attribute__((ext_vector_type(4))) unsigned v4u;
typedef __attribute__((ext_vector_type(4))) int      v4i;
typedef __attribute__((ext_vector_type(8))) int      v8i;

#define HAS_HW_BF8 (__has_builtin(__builtin_amdgcn_cvt_pk_bf8_f32) && \
                    __has_builtin(__builtin_amdgcn_cvt_pk_f32_bf8))

// Exact f32 -> fp8_e5m2 -> f32 round trip (round-to-nearest-even), software path.
// Inputs are pre-scaled into (-2, 2], so no overflow/Inf handling is needed.
__device__ __forceinline__ float rt_e5m2_sw(float v) {
  unsigned b   = __float_as_uint(v);
  unsigned lsb = (b >> 21) & 1u;
  float fn = __uint_as_float((b + 0xFFFFFu + lsb) & 0xFFE00000u);  // normal range RNE
  float fs = rintf(v * 65536.0f) * 0x1p-16f;                      // e5m2 subnormals
  return (__builtin_fabsf(v) >= 0x1p-14f) ? fn : fs;
}

// Quantize a float4 (hardware v_cvt_pk_bf8_f32 round trip when available).
__device__ __forceinline__ v4f rt_e5m2_x4(v4f a) {
#if HAS_HW_BF8
  short p01 = __builtin_amdgcn_cvt_pk_bf8_f32(a.x, a.y, 0, false);
  short p23 = __builtin_amdgcn_cvt_pk_bf8_f32(a.z, a.w, 0, false);
  v2f f01 = __builtin_amdgcn_cvt_pk_f32_bf8((int)(unsigned short)p01, false);
  v2f f23 = __builtin_amdgcn_cvt_pk_f32_bf8((int)(unsigned short)p23, false);
  v4f q;
  q.x = f01.x; q.y = f01.y; q.z = f23.x; q.w = f23.y;
  return q;
#else
  v4f q;
  q.x = rt_e5m2_sw(a.x); q.y = rt_e5m2_sw(a.y);
  q.z = rt_e5m2_sw(a.z); q.w = rt_e5m2_sw(a.w);
  return q;
#endif
}

__global__ __launch_bounds__(256) void quant_e5m2_tile(
    const float* __restrict__ x, float* __restrict__ outq, float* __restrict__ outs) {
  __shared__ __align__(16) float tile[TILE * TILE];  // 64 KB staging buffer
  __shared__ unsigned s_bits;                        // tile absmax as uint bits

  const int tcol = blockIdx.x;
  const int trow = blockIdx.y;
  const int tid  = threadIdx.x;
  const size_t gbase = (size_t)trow * TILE * GN + (size_t)tcol * TILE;

  if (tid == 0) s_bits = 0u;

#if __has_builtin(__builtin_amdgcn_tensor_load_to_lds)
  if (tid < 32) {  // wave 0 issues one TDM descriptor for the whole 64 KB tile
    unsigned lds_off = (unsigned)(uintptr_t)&tile[0];  // LDS byte offset (addr[31:0])
    uint64_t ga = (uint64_t)(uintptr_t)(x + gbase);    // global byte address of tile
    // D# group 0 (ISA 08 §8.3): count=1 | lds_addr | global_addr[56:0] | type=2
    v4u g0 = { 1u,
               lds_off,
               (unsigned)(ga & 0xFFFFFFFFu),
               (unsigned)((ga >> 32) & 0x01FFFFFFu) | (2u << 30) };
    // D# group 1 (§8.4): wg_mask=0, data_size=2 (4B), no barrier/iterate/pad,
    // tensor_dim0=8192, tensor_dim1=8192, tile_dim0=128, tile_dim1=128,
    // tile_dim2=0, tensor_dim0_stride=8192, tensor_dim1_stride=0.
    v8i g1 = { (int)(2u << 16),
               (int)(GN << 16),     // bits[63:48]   = tensor_dim0[15:0]
               (int)(GN << 16),     // bits[95:80]   = tensor_dim1[15:0]
               (int)(TILE << 16),   // bits[127:112] = tile_dim0
               (int)TILE,           // bits[143:128] = tile_dim1 ; tile_dim2=0
               (int)GN,             // bits[191:160] = tensor_dim0_stride[31:0]
               0, 0 };
    v4i z4 = {0, 0, 0, 0};          // groups 2/3 unused (2D tensor)
#if __has_include(<hip/amd_detail/amd_gfx1250_TDM.h>)
    v8i z8 = {0, 0, 0, 0, 0, 0, 0, 0};
    __builtin_amdgcn_tensor_load_to_lds(g0, g1, z4, z4, z8, 0);   // clang-23 6-arg form
#else
    __builtin_amdgcn_tensor_load_to_lds(g0, g1, z4, z4, 0);       // ROCm 7.2 5-arg form
#endif
#if __has_builtin(__builtin_amdgcn_s_wait_tensorcnt)
    __builtin_amdgcn_s_wait_tensorcnt(0);
#else
    asm volatile("s_wait_tensorcnt 0x0" ::: "memory");
#endif
  }
#else
  // Fallback path: cooperative b128 global->LDS copy (one full-wave row per step)
  for (int i = 0; i < 16; ++i) {
    int vec = tid + i * 256;
    int r = vec >> 5, c = (vec & 31) << 2;
    *(v4f*)&tile[r * TILE + c] = *(const v4f*)&x[gbase + (size_t)r * GN + c];
  }
#endif
  __syncthreads();

  // Single LDS read: pull this thread's 64 elements into registers and
  // accumulate the absmax on the fly (abs folds into v_max3 src modifiers).
  v4f vals[16];
  float m = 0.0f;
#pragma unroll
  for (int i = 0; i < 16; ++i) {
    int vec = tid + i * 256;
    int r = vec >> 5, c = (vec & 31) << 2;
    v4f v = *(const v4f*)&tile[r * TILE + c];
    vals[i] = v;
    m = fmaxf(m, fmaxf(fmaxf(__builtin_fabsf(v.x), __builtin_fabsf(v.y)),
                       fmaxf(__builtin_fabsf(v.z), __builtin_fabsf(v.w))));
  }
#pragma unroll
  for (int off = 16; off > 0; off >>= 1)
    m = fmaxf(m, __shfl_xor(m, off, 32));
  if ((tid & 31) == 0) atomicMax(&s_bits, __float_as_uint(m));  // 8 atomics/tile
  __syncthreads();

  // e = ceil(log2(absmax)) computed exactly from the bit pattern (normal case)
  unsigned mb = s_bits;
  int e;
  if (mb == 0u)
    e = 0;
  else if (mb >= 0x00800000u)
    e = (int)(mb >> 23) - 127 + ((mb & 0x7FFFFFu) ? 1 : 0);
  else
    e = (int)ceilf(log2f(__uint_as_float(mb)));  // denormal absmax (never hit here)
  float inv = ldexpf(1.0f, MAXEXP - e);          // exact power-of-two inverse scale
  if (tid == 0) outs[trow * NTILES + tcol] = ldexpf(1.0f, e - MAXEXP);

  // Quantize from registers and stream out with non-temporal b128 stores
#pragma unroll
  for (int i = 0; i < 16; ++i) {
    int vec = tid + i * 256;
    int r = vec >> 5, c = (vec & 31) << 2;
    v4f a = vals[i];
    a.x *= inv; a.y *= inv; a.z *= inv; a.w *= inv;
    v4f q = rt_e5m2_x4(a);
    __builtin_nontemporal_store(q, (v4f*)&outq[gbase + (size_t)r * GN + c]);
  }
}

extern "C" void kernel_launch(void* const* d_in, const int* in_sizes, int n_in,
                              void* d_out, int out_size, void* d_ws, size_t ws_size,
                              hipStream_t stream) {
  (void)in_sizes; (void)n_in; (void)out_size; (void)d_ws; (void)ws_size;
  const float* x = (const float*)d_in[0];
  float* outq = (float*)d_out;                    // [8192*8192] quantized (f32 round-trip)
  float* outs = outq + (size_t)GN * GN;           // [64*64] scales, concatenated after
  dim3 grid(NTILES, NTILES);
  quant_e5m2_tile<<<grid, dim3(256), 0, stream>>>(x, outq, outs);
}